// SCLLoss_54872502173891
// MI455X (gfx1250) — compile-verified
//
#include <hip/hip_runtime.h>
#include <hip/hip_bf16.h>
#include <math.h>

typedef __attribute__((ext_vector_type(2))) float v2f;
typedef __attribute__((ext_vector_type(8))) float v8f;

#define NROW 4096   // N = 2B
#define DDIM 1024   // D
#define BM 128
#define BN 128
#define BK 16
#define LDT 20      // padded LDS row stride (floats): 16-byte aligned, bank-conflict free
#define NEG_INF (-1.0e30f)

// -------------------------------------------------------------------------
// Kernel 1: sim = R * R^T with V_WMMA_F32_16X16X4_F32 (fp32 matrix cores)
// R[r] = (r < 2048) ? zjs[r] : zis[r-2048]
// -------------------------------------------------------------------------
__device__ __forceinline__ const float* row_ptr(const float* zis, const float* zjs, int r) {
    return (r < (NROW / 2)) ? (zjs + (size_t)r * DDIM)
                            : (zis + (size_t)(r - NROW / 2) * DDIM);
}

__global__ __launch_bounds__(256) void sim_gemm(const float* __restrict__ zis,
                                                const float* __restrict__ zjs,
                                                float* __restrict__ sim) {
    __shared__ float sA[BM * LDT];
    __shared__ float sB[BN * LDT];

    const int tid  = threadIdx.x;
    const int lane = tid & 31;
    const int wid  = tid >> 5;          // 0..7
    const int wm   = wid >> 2;          // 0..1  (64-row band)
    const int wn   = wid & 3;           // 0..3  (32-col band)
    const int lrow = lane & 15;
    const int khalf = lane >> 4;        // 0/1: K pair select per ISA A/B layout

    const int rowBaseM = blockIdx.y * BM;
    const int rowBaseN = blockIdx.x * BN;

    v8f acc[4][2];
    #pragma unroll
    for (int mt = 0; mt < 4; ++mt)
        #pragma unroll
        for (int nt = 0; nt < 2; ++nt)
            acc[mt][nt] = (v8f){0.f, 0.f, 0.f, 0.f, 0.f, 0.f, 0.f, 0.f};

    for (int k0 = 0; k0 < DDIM; k0 += BK) {
        // ---- stage A/B tiles (each 128x16 fp32) into LDS: 2 float4 per thread each
        #pragma unroll
        for (int i = 0; i < 2; ++i) {
            int f  = tid + i * 256;          // float4 slot 0..511
            int r  = f >> 2;
            int c4 = (f & 3) << 2;
            const float* sa = row_ptr(zis, zjs, rowBaseM + r) + k0 + c4;
            *(float4*)&sA[r * LDT + c4] = *(const float4*)sa;
            const float* sb = row_ptr(zis, zjs, rowBaseN + r) + k0 + c4;
            *(float4*)&sB[r * LDT + c4] = *(const float4*)sb;
        }
        __syncthreads();

        // ---- 4 WMMA k-steps of 4 per staged BK=16
        #pragma unroll
        for (int kk = 0; kk < 4; ++kk) {
            const int ko = kk * 4 + khalf * 2;   // lanes 0-15: K={0,1}; 16-31: K={2,3}
            v2f a[4], b[2];
            #pragma unroll
            for (int mt = 0; mt < 4; ++mt)
                a[mt] = *(const v2f*)&sA[(wm * 64 + mt * 16 + lrow) * LDT + ko];
            #pragma unroll
            for (int nt = 0; nt < 2; ++nt)
                b[nt] = *(const v2f*)&sB[(wn * 32 + nt * 16 + lrow) * LDT + ko];
            #pragma unroll
            for (int mt = 0; mt < 4; ++mt)
                #pragma unroll
                for (int nt = 0; nt < 2; ++nt)
                    acc[mt][nt] = __builtin_amdgcn_wmma_f32_16x16x4_f32(
                        false, a[mt], false, b[nt], (short)0, acc[mt][nt], false, false);
        }
        __syncthreads();
    }

    // ---- write D: VGPR v, lanes 0-15 -> M=v, N=lane; lanes 16-31 -> M=v+8
    const int gm0 = rowBaseM + wm * 64;
    const int gn0 = rowBaseN + wn * 32;
    #pragma unroll
    for (int mt = 0; mt < 4; ++mt)
        #pragma unroll
        for (int nt = 0; nt < 2; ++nt) {
            const int gc = gn0 + nt * 16 + lrow;
            #pragma unroll
            for (int v = 0; v < 8; ++v) {
                const int gr = gm0 + mt * 16 + khalf * 8 + v;
                sim[(size_t)gr * NROW + gc] = acc[mt][nt][v];
            }
        }
}

// -------------------------------------------------------------------------
// Kernel 2: per-row negative max and logsumexp (mask != 0 => negative)
// -------------------------------------------------------------------------
__global__ __launch_bounds__(256) void neg_stats(const float* __restrict__ sim,
                                                 const int* __restrict__ mask,
                                                 float* __restrict__ negMax,
                                                 float* __restrict__ negLse) {
    const int row = blockIdx.x;
    const int tid = threadIdx.x;
    const float* srow = sim + (size_t)row * NROW;
    const int*   mrow = mask + (size_t)row * NROW;

    __shared__ float red[256];

    float m = NEG_INF;
    for (int c = tid; c < NROW; c += 256)
        if (mrow[c]) m = fmaxf(m, srow[c]);
    red[tid] = m;
    __syncthreads();
    for (int s = 128; s > 0; s >>= 1) {
        if (tid < s) red[tid] = fmaxf(red[tid], red[tid + s]);
        __syncthreads();
    }
    const float rmax = red[0];
    __syncthreads();

    float sum = 0.f;
    for (int c = tid; c < NROW; c += 256)
        if (mrow[c]) sum += __expf(srow[c] - rmax);
    red[tid] = sum;
    __syncthreads();
    for (int s = 128; s > 0; s >>= 1) {
        if (tid < s) red[tid] += red[tid + s];
        __syncthreads();
    }
    if (tid == 0) {
        negMax[row] = rmax;
        negLse[row] = (red[0] > 0.f) ? (__logf(red[0]) + rmax) : NEG_INF;
    }
}

// -------------------------------------------------------------------------
// Kernel 3: per-row mean loss/score over positives (mask==0 && c!=row)
// loss_ij = logaddexp(sim, neg_lse) - sim ; score_ij = sim >= neg_max
// -------------------------------------------------------------------------
__global__ __launch_bounds__(256) void pos_loss(const float* __restrict__ sim,
                                                const int* __restrict__ mask,
                                                const float* __restrict__ negMax,
                                                const float* __restrict__ negLse,
                                                float* __restrict__ rowLoss,
                                                float* __restrict__ rowScore) {
    const int row = blockIdx.x;
    const int tid = threadIdx.x;
    const float* srow = sim + (size_t)row * NROW;
    const int*   mrow = mask + (size_t)row * NROW;
    const float lse  = negLse[row];
    const float nmax = negMax[row];

    float lsum = 0.f, ssum = 0.f, cnt = 0.f;
    for (int c = tid; c < NROW; c += 256) {
        if (!mrow[c] && c != row) {
            const float s  = srow[c];
            const float mm = fmaxf(s, lse);
            lsum += __logf(__expf(s - mm) + __expf(lse - mm)) + mm - s;
            ssum += (s >= nmax) ? 1.f : 0.f;
            cnt  += 1.f;
        }
    }

    __shared__ float rl[256], rs[256], rc[256];
    rl[tid] = lsum; rs[tid] = ssum; rc[tid] = cnt;
    __syncthreads();
    for (int s = 128; s > 0; s >>= 1) {
        if (tid < s) { rl[tid] += rl[tid + s]; rs[tid] += rs[tid + s]; rc[tid] += rc[tid + s]; }
        __syncthreads();
    }
    if (tid == 0) {
        const float c = fmaxf(rc[0], 1.f);   // >=1 by construction
        rowLoss[row]  = rl[0] / c;
        rowScore[row] = rs[0] / c;
    }
}

// -------------------------------------------------------------------------
// Kernel 4: deterministic final mean over rows -> d_out[0]=loss, d_out[1]=score
// -------------------------------------------------------------------------
__global__ __launch_bounds__(256) void final_reduce(const float* __restrict__ rowLoss,
                                                    const float* __restrict__ rowScore,
                                                    float* __restrict__ out) {
    const int tid = threadIdx.x;
    float ls = 0.f, sc = 0.f;
    for (int r = tid; r < NROW; r += 256) { ls += rowLoss[r]; sc += rowScore[r]; }
    __shared__ float rl[256], rs[256];
    rl[tid] = ls; rs[tid] = sc;
    __syncthreads();
    for (int s = 128; s > 0; s >>= 1) {
        if (tid < s) { rl[tid] += rl[tid + s]; rs[tid] += rs[tid + s]; }
        __syncthreads();
    }
    if (tid == 0) {
        out[0] = rl[0] / (float)NROW;
        out[1] = rs[0] / (float)NROW;
    }
}

// -------------------------------------------------------------------------
extern "C" void kernel_launch(void* const* d_in, const int* in_sizes, int n_in,
                              void* d_out, int out_size, void* d_ws, size_t ws_size,
                              hipStream_t stream) {
    const float* zis  = (const float*)d_in[0];   // [2048,1024] fp32
    const float* zjs  = (const float*)d_in[1];   // [2048,1024] fp32
    const int*   mask = (const int*)d_in[2];     // [4096,4096] int32

    float* sim      = (float*)d_ws;                       // 4096*4096 fp32 = 64 MB
    float* negMax   = sim + (size_t)NROW * NROW;          // [4096]
    float* negLse   = negMax + NROW;                      // [4096]
    float* rowLoss  = negLse + NROW;                      // [4096]
    float* rowScore = rowLoss + NROW;                     // [4096]

    dim3 ggrid(NROW / BN, NROW / BM);                     // 32 x 32
    sim_gemm<<<ggrid, 256, 0, stream>>>(zis, zjs, sim);
    neg_stats<<<NROW, 256, 0, stream>>>(sim, mask, negMax, negLse);
    pos_loss<<<NROW, 256, 0, stream>>>(sim, mask, negMax, negLse, rowLoss, rowScore);
    final_reduce<<<1, 256, 0, stream>>>(rowLoss, rowScore, (float*)d_out);
}